// MoEWrapper_10393820857166
// MI455X (gfx1250) — compile-verified
//
#include <hip/hip_runtime.h>
#include <hip/hip_bf16.h>

// MoE top-2: sparse dispatch (34.4 GFLOP f32) instead of dense (137 GFLOP).
// f32 tensor path: V_WMMA_F32_16X16X4_F32 (full f32 fidelity, reference is f32).
// v3: double-buffered software pipeline -
//   A tile: GLOBAL_LOAD_ASYNC_TO_LDS_B128 for chunk k+1 in flight during
//           compute of chunk k (ASYNCcnt), alternate LDS buffer.
//   B tile: global->VGPR loads for chunk k+1 issued before compute of chunk k
//           (loadcnt-hidden), committed to LDS at the top of the next iteration.
//   One s_barrier per K-chunk. Dynamic LDS (71KB; WGP allows up to 320KB).

static constexpr int cB    = 8192;
static constexpr int cDIN  = 1024;
static constexpr int cDOUT = 1024;
static constexpr int cE    = 8;
static constexpr int cRH   = 128;

static constexpr int MT = 128;   // tokens per block tile
static constexpr int NT = 128;   // output cols per block tile
static constexpr int KC = 32;    // K chunk staged in LDS

static constexpr int AS_STRIDE = 36; // words; 144B rows -> every 16B seg aligned
static constexpr int BS_STRIDE = 34; // words; even (b64-aligned k-pairs)

static constexpr int AS_BUF_WORDS = MT * AS_STRIDE;      // 4608
static constexpr int BS_BUF_WORDS = NT * BS_STRIDE;      // 4352
static constexpr size_t GEMM_LDS_BYTES =
    (size_t)(2 * AS_BUF_WORDS + 2 * BS_BUF_WORDS) * 4 + MT * 8;  // 72704

typedef float v2f __attribute__((ext_vector_type(2)));
typedef float v8f __attribute__((ext_vector_type(8)));

// ---------------------------------------------------------------------------
// Kernel 1: router. One block per token (128 threads = RH).
// ---------------------------------------------------------------------------
__global__ __launch_bounds__(128) void router_kernel(
    const float* __restrict__ x,
    const float* __restrict__ rw1, const float* __restrict__ rb1,
    const float* __restrict__ rw2, const float* __restrict__ rb2,
    const float* __restrict__ eb,
    int* __restrict__ slot_idx, float* __restrict__ slot_wgt,
    float* __restrict__ out)
{
    __shared__ float xsh[cDIN];
    __shared__ float hsh[cRH];
    __shared__ float logit[cE];
    __shared__ int   sIdx[2];
    __shared__ float sW[2];

    const int b = blockIdx.x;
    const int t = threadIdx.x;

    for (int i = t; i < cDIN; i += 128) xsh[i] = x[(size_t)b * cDIN + i];
    __syncthreads();

    float acc = rb1[t];
    for (int d = 0; d < cDIN; ++d)
        acc = fmaf(xsh[d], rw1[d * cRH + t], acc);
    hsh[t] = tanhf(acc);
    __syncthreads();

    if (t < cE) {
        float l = rb2[t];
        for (int r = 0; r < cRH; ++r)
            l = fmaf(hsh[r], rw2[r * cE + t], l);
        logit[t] = l;
    }
    __syncthreads();

    if (t == 0) {
        float m = logit[0];
        for (int e = 1; e < cE; ++e) m = fmaxf(m, logit[e]);
        float p[cE];
        float s = 0.f;
        for (int e = 0; e < cE; ++e) { p[e] = __expf(logit[e] - m); s += p[e]; }
        const float inv = 1.f / s;
        for (int e = 0; e < cE; ++e) p[e] *= inv;
        int i0 = 0;
        for (int e = 1; e < cE; ++e) if (p[e] > p[i0]) i0 = e;
        int i1 = (i0 == 0) ? 1 : 0;
        for (int e = 0; e < cE; ++e)
            if (e != i0 && p[e] > p[i1]) i1 = e;
        const float w0 = p[i0], w1 = p[i1];
        const float e1 = __expf(w1 - w0);           // w0 >= w1
        const float nw0 = 1.f / (1.f + e1);
        const float nw1 = e1 / (1.f + e1);
        slot_idx[b * 2 + 0] = i0;  slot_idx[b * 2 + 1] = i1;
        slot_wgt[b * 2 + 0] = nw0; slot_wgt[b * 2 + 1] = nw1;
        sIdx[0] = i0; sIdx[1] = i1; sW[0] = nw0; sW[1] = nw1;
    }
    __syncthreads();

    const int   e0 = sIdx[0], e1i = sIdx[1];
    const float w0 = sW[0],   w1  = sW[1];
    for (int j = t; j < cDOUT; j += 128)
        out[(size_t)b * cDOUT + j] = w0 * eb[e0 * cDOUT + j] + w1 * eb[e1i * cDOUT + j];
}

// ---------------------------------------------------------------------------
// Kernel 2: deterministic stable partition into per-(slot, expert) lists.
// ---------------------------------------------------------------------------
__global__ __launch_bounds__(256) void partition_kernel(
    const int* __restrict__ slot_idx, const float* __restrict__ slot_wgt,
    int* __restrict__ tok_list, float* __restrict__ wgt_list,
    int* __restrict__ cnt, int* __restrict__ off)
{
    __shared__ int c[256][16];
    __shared__ int tot[16];
    __shared__ int base[16];

    const int t = threadIdx.x;
    const int TPT = cB / 256;

    int local[16];
    for (int i = 0; i < 16; ++i) local[i] = 0;
    for (int s = 0; s < 2; ++s)
        for (int i = 0; i < TPT; ++i) {
            const int b = t * TPT + i;
            const int e = slot_idx[b * 2 + s];
            local[s * 8 + e]++;
        }
    for (int i = 0; i < 16; ++i) c[t][i] = local[i];
    __syncthreads();

    if (t < 16) {
        int run = 0;
        for (int u = 0; u < 256; ++u) { const int tmp = c[u][t]; c[u][t] = run; run += tmp; }
        tot[t] = run;
    }
    __syncthreads();

    if (t < 2) {
        int run = 0;
        for (int e = 0; e < 8; ++e) {
            base[t * 8 + e] = run;
            const int v = tot[t * 8 + e];
            cnt[t * 8 + e] = v;
            off[t * 8 + e] = run;
            run += v;
        }
    }
    __syncthreads();

    for (int i = 0; i < 16; ++i) local[i] = 0;
    for (int s = 0; s < 2; ++s)
        for (int i = 0; i < TPT; ++i) {
            const int b  = t * TPT + i;
            const int e  = slot_idx[b * 2 + s];
            const int se = s * 8 + e;
            const int pos = base[se] + c[t][se] + local[se]++;
            tok_list[s * cB + pos] = b;
            wgt_list[s * cB + pos] = slot_wgt[b * 2 + s];
        }
}

// ---------------------------------------------------------------------------
// Kernel 3: gathered expert GEMM, f32 WMMA, double-buffered async pipeline.
// grid = (cB/MT, cDOUT/NT, cE). Block = 256 threads = 8 wave32.
// Wave w: rows mSub*32..+31 (two 16-row frags) x cols nSub*64 (4 frags)
// -> 8 accumulators, 8 WMMA per 6 ds_load_b64 (merged into ds_load_2addr_b64).
// ---------------------------------------------------------------------------
__global__ __launch_bounds__(256) void expert_gemm(
    const float* __restrict__ x, const float* __restrict__ ew,
    const int* __restrict__ tok_list, const float* __restrict__ wgt_list,
    const int* __restrict__ cnt, const int* __restrict__ off,
    float* __restrict__ out, int slot)
{
    extern __shared__ __align__(16) char dynsmem[];
    float* AsBase = (float*)dynsmem;                                   // [2][MT][36]
    float* BsBase = (float*)(dynsmem + (size_t)2 * AS_BUF_WORDS * 4);  // [2][NT][34]
    int*   tokId  = (int*)(dynsmem + (size_t)(2 * AS_BUF_WORDS + 2 * BS_BUF_WORDS) * 4);
    float* tokW   = (float*)(tokId + MT);

    const int e    = blockIdx.z;
    const int nTok = cnt[slot * cE + e];
    const int m0   = blockIdx.x * MT;
    if (m0 >= nTok) return;
    const int mCount = (nTok - m0 < MT) ? (nTok - m0) : MT;
    const int n0 = blockIdx.y * NT;

    const int t = threadIdx.x;
    const int lbase = off[slot * cE + e] + m0;
    const int*   tl = tok_list + slot * cB + lbase;
    const float* wl = wgt_list + slot * cB + lbase;

    if (t < MT) {
        if (t < mCount) { tokId[t] = tl[t]; tokW[t] = wl[t]; }
        else            { tokId[t] = 0;     tokW[t] = 0.f; }   // pad rows: masked at store
    }
    __syncthreads();

    const int lane = t & 31;
    const int wv   = t >> 5;
    const int mSub = wv & 3;          // 0..3 -> 32-row band
    const int nSub = wv >> 2;         // 0..1 -> 64-col band
    const int half = lane >> 4;       // selects K pair {k,k+1} vs {k+2,k+3}
    const int l15  = lane & 15;

    v8f acc0[4] = {};
    v8f acc1[4] = {};

    const float* ewe = ew + (size_t)e * cDIN * cDOUT;
    const uint64_t xg = (uint64_t)(uintptr_t)x;

    const int myTok[4] = { tokId[(0 * 256 + t) >> 3], tokId[(1 * 256 + t) >> 3],
                           tokId[(2 * 256 + t) >> 3], tokId[(3 * 256 + t) >> 3] };
    const int brKr = -1; // (unused marker)

    // ---- prologue: A chunk 0 async -> buffer 0; B chunk 0 -> registers.
    float breg[16];
    {
        const uint32_t asOff = (uint32_t)(uintptr_t)AsBase;  // buffer 0
        #pragma unroll
        for (int i = 0; i < 4; ++i) {
            const int q = i * 256 + t;               // 1024 x 16B segments
            const int r = q >> 3, seg = q & 7;
            const uint64_t ga = xg + (((uint64_t)myTok[i] * cDIN) << 2) + ((uint64_t)seg << 4);
            const uint32_t la = asOff + (uint32_t)(r * (AS_STRIDE * 4) + seg * 16);
            asm volatile("global_load_async_to_lds_b128 %0, %1, off"
                         :: "v"(la), "v"(ga) : "memory");
        }
        #pragma unroll
        for (int i = 0; i < 16; ++i) {
            const int idx = i * 256 + t;
            const int kr = idx >> 7, nc = idx & 127;
            breg[i] = ewe[(size_t)kr * cDOUT + n0 + nc];
        }
    }

    int buf = 0;
    for (int kc = 0; kc < cDIN; kc += KC) {
        float* Ab = AsBase + buf * AS_BUF_WORDS;
        float* Bb = BsBase + buf * BS_BUF_WORDS;

        // commit staged B registers for chunk kc -> Bs[buf] (transposed [n][k])
        #pragma unroll
        for (int i = 0; i < 16; ++i) {
            const int idx = i * 256 + t;
            const int kr = idx >> 7, nc = idx & 127;
            Bb[nc * BS_STRIDE + kr] = breg[i];
        }
        asm volatile("s_wait_asynccnt 0" ::: "memory");  // A chunk kc landed
        __syncthreads();  // chunk kc visible; buffers of chunk kc-1 now free

        // ---- issue chunk kc+KC staging into the alternate buffer (overlaps compute)
        const int kn = kc + KC;
        if (kn < cDIN) {
            const uint32_t asOff = (uint32_t)(uintptr_t)(AsBase + (buf ^ 1) * AS_BUF_WORDS);
            #pragma unroll
            for (int i = 0; i < 4; ++i) {
                const int q = i * 256 + t;
                const int r = q >> 3, seg = q & 7;
                const uint64_t ga = xg + (((uint64_t)myTok[i] * cDIN + kn) << 2) + ((uint64_t)seg << 4);
                const uint32_t la = asOff + (uint32_t)(r * (AS_STRIDE * 4) + seg * 16);
                asm volatile("global_load_async_to_lds_b128 %0, %1, off"
                             :: "v"(la), "v"(ga) : "memory");
            }
            #pragma unroll
            for (int i = 0; i < 16; ++i) {
                const int idx = i * 256 + t;
                const int kr = idx >> 7, nc = idx & 127;
                breg[i] = ewe[(size_t)(kn + kr) * cDOUT + n0 + nc];
            }
            // prefetch the chunk after next into L2
            if (kn + KC < cDIN) {
                const int kr = t >> 7, nc = t & 127;
                __builtin_prefetch(&ewe[(size_t)(kn + KC + kr) * cDOUT + n0 + nc], 0, 0);
            }
        }

        // ---- compute chunk kc from As[buf]/Bs[buf]
        #pragma unroll
        for (int kk = 0; kk < KC; kk += 4) {
            const int kh = kk + 2 * half;
            const v2f a0 = *(const v2f*)&Ab[(mSub * 32 + l15) * AS_STRIDE + kh];
            const v2f a1 = *(const v2f*)&Ab[(mSub * 32 + 16 + l15) * AS_STRIDE + kh];
            #pragma unroll
            for (int j = 0; j < 4; ++j) {
                const v2f bv = *(const v2f*)&Bb[(nSub * 64 + j * 16 + l15) * BS_STRIDE + kh];
                acc0[j] = __builtin_amdgcn_wmma_f32_16x16x4_f32(
                    false, a0, false, bv, (short)0, acc0[j], false, false);
                acc1[j] = __builtin_amdgcn_wmma_f32_16x16x4_f32(
                    false, a1, false, bv, (short)0, acc1[j], false, false);
            }
        }
        buf ^= 1;
    }
    (void)brKr;

    // epilogue: row-scale by routing weight, accumulate into out.
    // C/D layout: lanes 0-15 VGPR i -> M=i; lanes 16-31 VGPR i -> M=i+8; N=l15.
    #pragma unroll
    for (int f = 0; f < 2; ++f) {
        #pragma unroll
        for (int j = 0; j < 4; ++j) {
            #pragma unroll
            for (int i = 0; i < 8; ++i) {
                const int ml = mSub * 32 + f * 16 + i + 8 * half;
                if (ml < mCount) {
                    const int   tok = tokId[ml];
                    const float w   = tokW[ml];
                    const int   col = n0 + nSub * 64 + j * 16 + l15;
                    float* p = &out[(size_t)tok * cDOUT + col];
                    const float v = (f == 0) ? acc0[j][i] : acc1[j][i];
                    *p += w * v;
                }
            }
        }
    }
}

// ---------------------------------------------------------------------------
extern "C" void kernel_launch(void* const* d_in, const int* in_sizes, int n_in,
                              void* d_out, int out_size, void* d_ws, size_t ws_size,
                              hipStream_t stream) {
    (void)in_sizes; (void)n_in; (void)out_size; (void)ws_size;
    const float* x   = (const float*)d_in[0];
    const float* rw1 = (const float*)d_in[1];
    const float* rb1 = (const float*)d_in[2];
    const float* rw2 = (const float*)d_in[3];
    const float* rb2 = (const float*)d_in[4];
    const float* ew  = (const float*)d_in[5];
    const float* eb  = (const float*)d_in[6];
    float* out = (float*)d_out;

    char* w = (char*)d_ws;
    int*   slot_idx = (int*)w;    w += (size_t)cB * 2 * sizeof(int);
    float* slot_wgt = (float*)w;  w += (size_t)cB * 2 * sizeof(float);
    int*   tok_list = (int*)w;    w += (size_t)cB * 2 * sizeof(int);
    float* wgt_list = (float*)w;  w += (size_t)cB * 2 * sizeof(float);
    int*   cnt      = (int*)w;    w += 16 * sizeof(int);
    int*   off      = (int*)w;    w += 16 * sizeof(int);

    router_kernel<<<cB, 128, 0, stream>>>(x, rw1, rb1, rw2, rb2, eb,
                                          slot_idx, slot_wgt, out);
    partition_kernel<<<1, 256, 0, stream>>>(slot_idx, slot_wgt,
                                            tok_list, wgt_list, cnt, off);
    dim3 grid(cB / MT, cDOUT / NT, cE);
    expert_gemm<<<grid, 256, GEMM_LDS_BYTES, stream>>>(
        x, ew, tok_list, wgt_list, cnt, off, out, 0);
    expert_gemm<<<grid, 256, GEMM_LDS_BYTES, stream>>>(
        x, ew, tok_list, wgt_list, cnt, off, out, 1);
}